// CritiGraph_35579509080218
// MI455X (gfx1250) — compile-verified
//
#include <hip/hip_runtime.h>
#include <hip/hip_bf16.h>

typedef __attribute__((ext_vector_type(2))) float v2f;
typedef __attribute__((ext_vector_type(8))) float v8f;

namespace {
constexpr int TB = 8;    // B
constexpr int TT = 32;   // T
constexpr int TD = 8;    // D (TP)
constexpr int TC = 513;  // candidates
constexpr int NG = 33;   // ceil(C/16) candidate groups
}

__global__ __launch_bounds__(256)
void CritiGraph_wmma_kernel(const int* __restrict__ sta,       // (B,T,D) i32
                            const int* __restrict__ cnc,       // (B,T,C,D) i32
                            const float* __restrict__ logits,  // (B,T,T) f32
                            const int* __restrict__ lg,        // (B,T) i32
                            const unsigned char* __restrict__ mask, // (B,T,T) bool
                            float* __restrict__ out)           // 2048 sel + 2048 loss
{
  __shared__ int   sraw[TT][TD];     // raw station coords (for sign xor)
  __shared__ int   sabs[TT][TD];     // |station|
  __shared__ float dspS[TT][TD];     // dist(sta_i, sta_j) per dim
  __shared__ float offv[TT][TD];     // dsps_j - dsp_jd   (exact)
  __shared__ float dsps[TT];
  __shared__ float llog[TT];
  __shared__ float lmsk[TT];
  __shared__ float resL[TD][NG * 2];
  __shared__ int   resC[TD][NG * 2];

  const int tid  = threadIdx.x;
  const int blk  = blockIdx.x;      // b*T + i
  const int b    = blk >> 5;
  const int irow = blk & 31;

  // ---- stage per-(b,i) invariants ----
  {
    const int j = tid >> 3;
    const int d = tid & 7;
    const int s2 = sta[(b * TT + j) * TD + d];
    const int s1 = sta[(b * TT + irow) * TD + d];
    sraw[j][d] = s2;
    const int a2 = (s2 < 0) ? -s2 : s2;
    sabs[j][d] = a2;
    const int a1 = (s1 < 0) ? -s1 : s1;
    // 1 - lut[x] = (clz(x+1)-16)/16 ; argument >= 1 so __builtin_clz is safe
    const int cl = (int)__builtin_clz((unsigned)((a1 ^ a2) + 1));
    const float fm = ((s1 ^ s2) < 0) ? -0.0625f : 0.0625f;  // sign agreement
    dspS[j][d] = (float)(cl - 16) * fm;       // exact multiple of 1/16
    if (tid < TT) {
      llog[tid] = logits[(b * TT + irow) * TT + tid];
      lmsk[tid] = mask[(b * TT + irow) * TT + tid] ? 1.0f : 0.0f;
    }
  }
  __syncthreads();
  if (tid < TT) {
    float s = 0.0f;
#pragma unroll
    for (int d = 0; d < TD; ++d) s += dspS[tid][d];   // exact (1/16 grid)
    dsps[tid] = s;
  }
  __syncthreads();
  {
    const int j = tid >> 3;
    const int d = tid & 7;
    offv[j][d] = dsps[j] - dspS[j][d];                // exact
  }
  __syncthreads();

  // ---- main sweep: wave = dim d; 33 groups of 16 candidates ----
  const int d    = tid >> 5;
  const int lane = tid & 31;
  const int mrow = lane & 15;                 // A-matrix row M = candidate in group
  const int koff = (lane < 16) ? 0 : 2;       // A-layout: lanes 0-15 hold K=0,1; 16-31 hold K=2,3
  const v2f bones = {1.0f, 1.0f};             // B = ones(4x16) -> row-sum reduction
  const int baseC = (b * TT + irow) * TC;

  for (int g = 0; g < NG; ++g) {
    int c = g * 16 + mrow;
    if (c > TC - 1) c = TC - 1;               // clamp pad lanes (masked at argmin)
    const int cv  = cnc[(baseC + c) * TD + d];
    const int cab = (cv < 0) ? -cv : cv;
    v8f acc = {0.f, 0.f, 0.f, 0.f, 0.f, 0.f, 0.f, 0.f};
#pragma unroll
    for (int q = 0; q < 8; ++q) {             // 8 chunks x K=4 = 32 stations
      v2f a;
#pragma unroll
      for (int u = 0; u < 2; ++u) {
        const int j  = q * 4 + koff + u;
        const int sj = sraw[j][d];
        // clz argument is >= 1 by construction -> no clamp emitted
        const int cl = (int)__builtin_clz((unsigned)((sabs[j][d] ^ cab) + 1));
        const float fm = ((sj ^ cv) < 0) ? -0.0625f : 0.0625f;
        float t = fmaf((float)(cl - 16), fm, offv[j][d]); // exact
        t = fmaf(t, 0.125f, -llog[j]);                    // exact *0.125, 1 rounding
        t = fabsf(t * lmsk[j]);
        if (u == 0) a.x = t; else a.y = t;
      }
      // D(16x16) += A(16x4) x ones(4x16) : j-reduction on the matrix pipe
      acc = __builtin_amdgcn_wmma_f32_16x16x4_f32(false, a, false, bones,
                                                  (short)0, acc, false, false);
    }
    // D layout: lane0 vgpr r -> M=r (cands g*16+0..7); lane16 -> M=8+r
    if ((lane & 15) == 0) {
      const int half  = lane >> 4;
      const int cbase = g * 16 + half * 8;
      float bl = 3.0e38f;
      int   bc = 0;
#pragma unroll
      for (int r = 0; r < 8; ++r) {
        const int cc = cbase + r;
        const float lv = acc[r];
        if (cc < TC && lv < bl) { bl = lv; bc = cc; }  // strict < = first-min
      }
      resL[d][g * 2 + half] = bl;
      resC[d][g * 2 + half] = bc;
    }
  }
  __syncthreads();

  // ---- final argmin per dim, ascending candidate order ----
  if (tid < TD) {
    float bl = 3.0e38f;
    int   bc = 0;
    for (int e = 0; e < NG * 2; ++e) {
      const float lv = resL[tid][e];
      if (lv < bl) { bl = lv; bc = resC[tid][e]; }
    }
    const float lgf = (float)lg[b * TT + irow];
    const int sel   = cnc[(baseC + bc) * TD + tid];
    const int oidx  = (b * TT + irow) * TD + tid;
    out[oidx] = (float)sel;                       // selected_locs
    out[TB * TT * TD + oidx] = bl / lgf;          // min_loss (exact /32)
  }
}

extern "C" void kernel_launch(void* const* d_in, const int* in_sizes, int n_in,
                              void* d_out, int out_size, void* d_ws, size_t ws_size,
                              hipStream_t stream) {
  const int*   sta    = (const int*)d_in[0];
  const int*   cnc    = (const int*)d_in[1];
  const float* logits = (const float*)d_in[2];
  const int*   lg     = (const int*)d_in[3];
  const unsigned char* mask = (const unsigned char*)d_in[4];
  // d_in[5] = lut: replaced analytically by clz (no dependent gather)
  float* out = (float*)d_out;
  CritiGraph_wmma_kernel<<<dim3(TB * TT), 256, 0, stream>>>(sta, cnc, logits, lg,
                                                            mask, out);
}